// CommonSeqTagModel_20409684591199
// MI455X (gfx1250) — compile-verified
//
#include <hip/hip_runtime.h>
#include <hip/hip_bf16.h>

// Problem constants (from the reference)
#define Bn 64
#define Sn 512
#define Hn 768
#define Tn 3
#define ROWS (Bn * Sn)      // 32768
#define TILES (ROWS / 16)   // 2048

typedef float v2f __attribute__((ext_vector_type(2)));
typedef float v8f __attribute__((ext_vector_type(8)));

// ---------------------------------------------------------------------------
// Kernel 1: emissions = relu(hidden @ W + b) using V_WMMA_F32_16X16X4_F32
// One wave per 16-row tile of (B*S, H) x (H, T->padded 16).
// A-frag (16x4 f32): lane L<16 holds (M=L, K=0/1), lane 16+L holds (M=L, K=2/3)
// B-frag (4x16 f32): lane L<16 holds (K=0/1, N=L), lane 16+L holds (K=2/3, N=L)
// C/D (16x16 f32):   VGPR r, lanes 0-15 -> M=r, lanes 16-31 -> M=8+r
// ---------------------------------------------------------------------------
__global__ __launch_bounds__(256) void emis_wmma_kernel(
    const float* __restrict__ hidden,   // [B*S, H]
    const float* __restrict__ W,        // [H, T]
    const float* __restrict__ bias,     // [T]
    float* __restrict__ out)            // [B*S, T]
{
    __shared__ float sW[Hn * Tn];       // 9216 B
    const int tid = threadIdx.x;
    #pragma unroll
    for (int i = tid; i < Hn * Tn; i += 256) sW[i] = W[i];
    __syncthreads();

    const int wave = tid >> 5;
    const int lane = tid & 31;
    const int tile = blockIdx.x * 8 + wave;

    const int rloc = lane & 15;          // M for A-frag, N for B-frag
    const int kb   = (lane >> 4) << 1;   // 0 or 2 (K pair selector)
    const float nscale = (rloc < Tn) ? 1.0f : 0.0f;  // zero-pad B beyond T cols
    const int   nc     = (rloc < Tn) ? rloc : 0;

    const float* pA = hidden + (size_t)(tile * 16 + rloc) * Hn + kb;

    v8f acc = {0.f, 0.f, 0.f, 0.f, 0.f, 0.f, 0.f, 0.f};

    // H=768 -> 12 outer chunks of 64 K (one 256B span per row), 16 WMMA each
    for (int k0 = 0; k0 < Hn; k0 += 64) {
        if (k0 + 128 < Hn)
            __builtin_prefetch(pA + k0 + 128, 0, 3);   // global_prefetch_b8
        #pragma unroll
        for (int k4 = 0; k4 < 64; k4 += 4) {
            const int kk = k0 + k4 + kb;
            v2f a = *(const v2f*)(pA + k0 + k4);       // global_load_b64
            v2f bf;
            bf.x = sW[kk * Tn + nc] * nscale;          // ds_load_b32
            bf.y = sW[(kk + 1) * Tn + nc] * nscale;    // ds_load_b32
            acc = __builtin_amdgcn_wmma_f32_16x16x4_f32(
                false, a, false, bf, (short)0, acc, false, false);
        }
    }

    // Epilogue: bias + relu, store only the T=3 valid columns
    if (rloc < Tn) {
        const float bb = bias[rloc];
        const int mbase = (lane >> 4) * 8;
        #pragma unroll
        for (int r = 0; r < 8; ++r) {
            float v = acc[r] + bb;
            v = v > 0.0f ? v : 0.0f;
            out[(size_t)(tile * 16 + mbase + r) * Tn + rloc] = v;
        }
    }
}

// ---------------------------------------------------------------------------
// Kernel 2: CRF NLL, one wave (32 lanes) per batch element.
//  - whole wave: masked numerator partial sums + length (shfl reduction)
//  - lanes 0..2: parallel 3-tag forward logsumexp recursion via shfl
// ---------------------------------------------------------------------------
__device__ __forceinline__ float lse3(float x, float y, float z) {
    const float L2E = 1.44269504088896340736f;
    const float LN2 = 0.69314718055994530942f;
    float m = fmaxf(x, fmaxf(y, z));
    float s = __builtin_amdgcn_exp2f((x - m) * L2E)
            + __builtin_amdgcn_exp2f((y - m) * L2E)
            + __builtin_amdgcn_exp2f((z - m) * L2E);
    return m + LN2 * __builtin_amdgcn_logf(s);
}

__global__ __launch_bounds__(32) void crf_kernel(
    const float* __restrict__ emis,     // [B, S, T] (post-relu, from kernel 1)
    const int*   __restrict__ labels,   // [B, S]
    const int*   __restrict__ mask,     // [B, S]
    const float* __restrict__ trans,    // [T, T]
    const float* __restrict__ start,    // [T]
    const float* __restrict__ endt,     // [T]
    float*       __restrict__ partial)  // [B] per-batch (logZ - num)
{
    const int b    = blockIdx.x;
    const int lane = threadIdx.x;

    const float* eb = emis + (size_t)b * Sn * Tn;
    const int*   lb = labels + b * Sn;
    const int*   mb = mask + b * Sn;

    float tr[9];
    #pragma unroll
    for (int i = 0; i < 9; ++i) tr[i] = trans[i];

    // ---- numerator: parallel over timesteps ----
    float stepsum = 0.0f;
    int   lensum  = 0;
    for (int t = lane; t < Sn; t += 32) {
        const int m = mb[t];
        lensum += m;
        if (t >= 1) {
            const int prev = lb[t - 1];
            const int cur  = lb[t];
            const float st = tr[prev * 3 + cur] + eb[t * Tn + cur];
            stepsum += (m > 0) ? st : 0.0f;
        }
    }
    #pragma unroll
    for (int off = 16; off > 0; off >>= 1) {
        stepsum += __shfl_down(stepsum, off);
        lensum  += __shfl_down(lensum, off);
    }
    stepsum = __shfl(stepsum, 0);
    lensum  = __shfl(lensum, 0);

    // ---- forward scan: lanes 0..2 each own one tag's alpha ----
    const int j = (lane < Tn) ? lane : 0;
    const float tr0 = tr[0 * 3 + j];
    const float tr1 = tr[1 * 3 + j];
    const float tr2 = tr[2 * 3 + j];

    float a = start[j] + eb[j];   // alpha0[j]
    #pragma unroll 4
    for (int t = 1; t < Sn; ++t) {
        const float a0 = __shfl(a, 0);
        const float a1 = __shfl(a, 1);
        const float a2 = __shfl(a, 2);
        const float nxt = lse3(a0 + tr0, a1 + tr1, a2 + tr2) + eb[t * Tn + j];
        a = (mb[t] > 0) ? nxt : a;
    }
    a += endt[j];
    const float f0 = __shfl(a, 0);
    const float f1 = __shfl(a, 1);
    const float f2 = __shfl(a, 2);

    if (lane == 0) {
        const float logZ = lse3(f0, f1, f2);
        const int ft = lb[0];
        const int lt = lb[lensum - 1];
        const float num = start[ft] + eb[ft] + stepsum + endt[lt];
        partial[b] = logZ - num;
    }
}

// ---------------------------------------------------------------------------
// Kernel 3: deterministic fixed-order tree reduction of 64 per-batch values
// ---------------------------------------------------------------------------
__global__ __launch_bounds__(64) void reduce_kernel(
    const float* __restrict__ partial, float* __restrict__ out)
{
    __shared__ float s[64];
    const int t = threadIdx.x;
    s[t] = partial[t];
    __syncthreads();
    #pragma unroll
    for (int off = 32; off > 0; off >>= 1) {
        if (t < off) s[t] += s[t + off];
        __syncthreads();
    }
    if (t == 0) out[(size_t)ROWS * Tn] = s[0];
}

// ---------------------------------------------------------------------------
extern "C" void kernel_launch(void* const* d_in, const int* in_sizes, int n_in,
                              void* d_out, int out_size, void* d_ws, size_t ws_size,
                              hipStream_t stream)
{
    const float* hidden = (const float*)d_in[0];
    const int*   labels = (const int*)  d_in[1];
    const int*   maskp  = (const int*)  d_in[2];
    const float* W      = (const float*)d_in[3];
    const float* bias   = (const float*)d_in[4];
    const float* trans  = (const float*)d_in[5];
    const float* start  = (const float*)d_in[6];
    const float* endt   = (const float*)d_in[7];

    float* out     = (float*)d_out;           // [B*S*T] emissions + [1] nll
    float* partial = (float*)d_ws;            // 64 floats of scratch

    emis_wmma_kernel<<<TILES / 8, 256, 0, stream>>>(hidden, W, bias, out);
    crf_kernel<<<Bn, 32, 0, stream>>>(out, labels, maskp, trans, start, endt, partial);
    reduce_kernel<<<1, 64, 0, stream>>>(partial, out);
}